// ProdLayer_31696858644649
// MI455X (gfx1250) — compile-verified
//
#include <hip/hip_runtime.h>

// Segment-product over sorted segment ids (CSR-style), MI455X / gfx1250.
//
// out[s] = prod_{i: csr[i]==s} x[ptrs[i]],  empty segments -> 1.0
//
// One thread per output segment; csr is sorted, so each segment is a
// contiguous edge range found by binary search. Hierarchy:
//   1) two threads per block find the block's edge window [blo, bhi)
//   2) each thread binary-searches its key inside that ~1K-edge window
//   3) hi(s) == lo(s+1) comes from lane+1 via a wave32 shuffle
// Deterministic in-order per-segment product, no atomics, no init pass
// (empty segments naturally store 1.0).
//
// All values (ptrs < 4M, csr < 4M, E = 16M) are non-negative and fit in the
// low dword of their little-endian int64 storage -> 32-bit loads + 32-bit
// index math throughout.
//
// MI455X memory reasoning: x (16MB) and csr working set stay L2-resident
// (192MB L2); ptrs (128MB) is a read-once stream -> non-temporal loads.

#define BLOCK 256

__global__ void __launch_bounds__(BLOCK)
seg_prod_kernel(const float* __restrict__ x,
                const int* __restrict__ ptrs_lo,   // low dwords of int64 ptrs
                const int* __restrict__ csr_lo,    // low dwords of int64 csr
                float* __restrict__ out,
                int num_edges, int num_segments) {
  const int tid  = threadIdx.x;
  const int s    = blockIdx.x * BLOCK + tid;
  const int lane = tid & 31;
  const bool active = (s < num_segments);

  // ---- 1) block-level narrowing: edge window covering keys [s0, keyEnd] ----
  __shared__ int sRange[2];
  {
    const int s0     = blockIdx.x * BLOCK;
    const int keyEnd = min(s0 + BLOCK, num_segments);
    if (tid < 2) {
      const int k = (tid == 0) ? s0 : keyEnd;
      int lo = 0, hi = num_edges;
      while (lo < hi) {
        int mid = (lo + hi) >> 1;
        if (csr_lo[2 * mid] < k) lo = mid + 1; else hi = mid;
      }
      sRange[tid] = lo;
    }
  }
  __syncthreads();
  const int blo = sRange[0];
  const int bhi = sRange[1];

  // ---- 2) per-thread search inside the narrowed window ----
  // Inactive lanes search key == num_segments (all csr values are smaller,
  // so the result stays within [blo, bhi]); this is exactly the hi their
  // left neighbor needs, so all 32 lanes participate in the shuffle.
  const int key = active ? s : num_segments;
  int lo = blo, hi = bhi;
  while (lo < hi) {
    int mid = (lo + hi) >> 1;
    if (csr_lo[2 * mid] < key) lo = mid + 1; else hi = mid;
  }

  // Warm this thread's edge run while lane 31 does its extra search.
  if (lo < num_edges) __builtin_prefetch(&ptrs_lo[2 * lo], 0, 0);

  // ---- 3) hi(s) = lo(s+1) from lane+1; lane 31 searches itself ----
  const int nb = __shfl_down(lo, 1, 32);
  int hiIdx;
  if (lane == 31) {
    const int k2 = key + 1;
    int l2 = lo, h2 = bhi;   // lower_bound(key+1) in [lo, bhi]
    while (l2 < h2) {
      int mid = (l2 + h2) >> 1;
      if (csr_lo[2 * mid] < k2) l2 = mid + 1; else h2 = mid;
    }
    hiIdx = l2;
  } else {
    hiIdx = nb;
  }

  // ---- in-order product over the segment's edge run ----
  // ptrs is a read-once 128MB stream -> non-temporal; x gathers stay regular
  // (x is L2-resident at 16MB).
  float p = 1.0f;
  for (int i = lo; i < hiIdx; ++i) {
    int idx = __builtin_nontemporal_load(&ptrs_lo[2 * i]);
    p *= x[idx];
  }

  if (active) out[s] = p;
}

extern "C" void kernel_launch(void* const* d_in, const int* in_sizes, int n_in,
                              void* d_out, int out_size, void* d_ws, size_t ws_size,
                              hipStream_t stream) {
  (void)n_in; (void)d_ws; (void)ws_size;
  const float* x       = (const float*)d_in[0];
  const int*   ptrs_lo = (const int*)d_in[1];   // view int64 array as dwords
  const int*   csr_lo  = (const int*)d_in[2];   // view int64 array as dwords
  float*       out     = (float*)d_out;

  const int num_edges    = in_sizes[1];
  const int num_segments = out_size;  // == reference num_segments

  const int grid = (num_segments + BLOCK - 1) / BLOCK;
  seg_prod_kernel<<<grid, BLOCK, 0, stream>>>(x, ptrs_lo, csr_lo, out,
                                              num_edges, num_segments);
}